// meta_learning_77816217469351
// MI455X (gfx1250) — compile-verified
//
#include <hip/hip_runtime.h>
#include <math.h>

#define NSUP 2048
#define NQRY 4096
#define HDIM 256
#define NR   (NSUP + NQRY)
#define CG_ITERS 128

typedef __attribute__((ext_vector_type(2))) float v2f;
typedef __attribute__((ext_vector_type(8))) float v8f;
typedef __attribute__((ext_vector_type(4))) int v4i;
typedef __attribute__((address_space(1))) v4i as1_v4i;
typedef __attribute__((address_space(3))) v4i as3_v4i;

__device__ __forceinline__ v8f wmma_f32_16x16x4(v2f a, v2f b, v8f c) {
#if defined(__AMDGCN__) && __has_builtin(__builtin_amdgcn_wmma_f32_16x16x4_f32)
    return __builtin_amdgcn_wmma_f32_16x16x4_f32(false, a, false, b, (short)0, c,
                                                 false, false);
#else
    // host-pass stub (never executed on device)
    for (int i = 0; i < 8; ++i) c[i] += a[0] * b[0];
    return c;
#endif
}

// ---- gfx1250 async global->LDS copy (ASYNCcnt path), with safe fallback ----
__device__ __forceinline__ void async_copy_b128(float* lds, const float* gptr) {
#if defined(__AMDGCN__) && __has_builtin(__builtin_amdgcn_global_load_async_to_lds_b128) && \
    __has_builtin(__builtin_amdgcn_s_wait_asynccnt)
    __builtin_amdgcn_global_load_async_to_lds_b128(
        (as1_v4i*)(gptr), (as3_v4i*)(lds), 0, 0);
#else
    ((float4*)lds)[0] = ((const float4*)gptr)[0];
#endif
}
__device__ __forceinline__ void async_wait_all() {
#if defined(__AMDGCN__) && __has_builtin(__builtin_amdgcn_global_load_async_to_lds_b128) && \
    __has_builtin(__builtin_amdgcn_s_wait_asynccnt)
    __builtin_amdgcn_s_wait_asynccnt(0);
#endif
}

__device__ __forceinline__ float block_reduce_256(float v, float* sm) {
    int t = threadIdx.x;
    sm[t] = v;
    __syncthreads();
    for (int s = 128; s > 0; s >>= 1) {
        if (t < s) sm[t] += sm[t + s];
        __syncthreads();
    }
    float r = sm[0];
    __syncthreads();
    return r;
}

// hbar[j] = mean_i relu(S[i,:3] . Wz1[j,:3] + bz1[j]);  grid=HDIM, block=256
__global__ void k_colmean_h1(const float* __restrict__ S, const float* __restrict__ Wz1,
                             const float* __restrict__ bz1, float* __restrict__ hbar) {
    __shared__ float sm[256];
    int j = blockIdx.x;
    float w0 = Wz1[j * 3 + 0], w1 = Wz1[j * 3 + 1], w2 = Wz1[j * 3 + 2], b = bz1[j];
    float acc = 0.f;
    for (int i = threadIdx.x; i < NSUP; i += 256) {
        float v = fmaf(S[i * 3 + 0], w0, fmaf(S[i * 3 + 1], w1, fmaf(S[i * 3 + 2], w2, b)));
        acc += fmaxf(v, 0.f);
    }
    float s = block_reduce_256(acc, sm);
    if (threadIdx.x == 0) hbar[j] = s * (1.0f / (float)NSUP);
}

// out[j] = act( sum_k W[j*ld + off + k] * v[k] + bias[j] );  grid=HDIM, block=256
__global__ void k_matvec(const float* __restrict__ W, int ld, int off,
                         const float* __restrict__ v, const float* __restrict__ bias,
                         float* __restrict__ out, int kdim, int relu) {
    __shared__ float sm[256];
    int j = blockIdx.x;
    float acc = 0.f;
    for (int k = threadIdx.x; k < kdim; k += 256)
        acc += W[(size_t)j * ld + off + k] * v[k];
    float s = block_reduce_256(acc, sm);
    if (threadIdx.x == 0) {
        float r = s + bias[j];
        out[j] = relu ? fmaxf(r, 0.f) : r;
    }
}

// out[0] = w . v + b[0];  grid=1, block=256 (kdim == 256)
__global__ void k_dot_bias(const float* __restrict__ w, const float* __restrict__ v,
                           const float* __restrict__ b, float* __restrict__ out) {
    __shared__ float sm[256];
    float s = block_reduce_256(w[threadIdx.x] * v[threadIdx.x], sm);
    if (threadIdx.x == 0) out[0] = s + b[0];
}

// First hidden layer of mean/K nets (rank-2 + const) + mean head.
// grid=NR, block=256
__global__ void k_layer1(const float* __restrict__ S, const float* __restrict__ Qs,
                         const float* __restrict__ Wm1, const float* __restrict__ cm,
                         const float* __restrict__ Wk1, const float* __restrict__ ck,
                         const float* __restrict__ Wm2, const float* __restrict__ bm2,
                         float* __restrict__ Hk, float* __restrict__ meanv) {
    __shared__ float sm[256];
    int i = blockIdx.x, j = threadIdx.x;
    float x0, x1;
    if (i < NSUP) { x0 = S[i * 3 + 0]; x1 = S[i * 3 + 1]; }
    else          { int q = i - NSUP; x0 = Qs[q * 2 + 0]; x1 = Qs[q * 2 + 1]; }
    const int LD = HDIM + 2;
    float hm = fmaxf(fmaf(x0, Wm1[j * LD + 0], fmaf(x1, Wm1[j * LD + 1], cm[j])), 0.f);
    float hk = fmaxf(fmaf(x0, Wk1[j * LD + 0], fmaf(x1, Wk1[j * LD + 1], ck[j])), 0.f);
    Hk[(size_t)i * HDIM + j] = hk;
    float s = block_reduce_256(hm * Wm2[j], sm);
    if (j == 0) meanv[i] = s + bm2[0];
}

// Kall = Hk(NR x H) @ Wk2^T + bk2. One 16x64 strip per wave (A fragment reused x4).
// grid = (NR/16)*(HDIM/64)/8, block=256
__global__ void k_gemm_kall(const float* __restrict__ Hk, const float* __restrict__ Wk2,
                            const float* __restrict__ bk2, float* __restrict__ Kall) {
    int lane = threadIdx.x & 31;
    int wave = threadIdx.x >> 5;
    int strip = blockIdx.x * 8 + wave;
    const int NG = HDIM / 64;                 // 4 N-groups
    int tm = strip / NG, ng = strip % NG;

    int arow = tm * 16 + (lane & 15);
    int bcol = ng * 64 + (lane & 15);
    int khalf = (lane >> 4) * 2;
    const float* ap = Hk + (size_t)arow * HDIM + khalf;
    const float* bp = Wk2 + (size_t)bcol * HDIM + khalf;

    v8f acc0 = {}, acc1 = {}, acc2 = {}, acc3 = {};
    for (int k0 = 0; k0 < HDIM; k0 += 4) {
        v2f a = *(const v2f*)(ap + k0);
        v2f b0 = *(const v2f*)(bp + k0);
        v2f b1 = *(const v2f*)(bp + k0 + 16 * HDIM);
        v2f b2 = *(const v2f*)(bp + k0 + 32 * HDIM);
        v2f b3 = *(const v2f*)(bp + k0 + 48 * HDIM);
        acc0 = wmma_f32_16x16x4(a, b0, acc0);
        acc1 = wmma_f32_16x16x4(a, b1, acc1);
        acc2 = wmma_f32_16x16x4(a, b2, acc2);
        acc3 = wmma_f32_16x16x4(a, b3, acc3);
    }
    int mbase = tm * 16 + ((lane >> 4) << 3);
#pragma unroll
    for (int t = 0; t < 4; ++t) {
        v8f acc = (t == 0) ? acc0 : (t == 1) ? acc1 : (t == 2) ? acc2 : acc3;
        int n = ng * 64 + t * 16 + (lane & 15);
        float bias = bk2[n];
#pragma unroll
        for (int r = 0; r < 8; ++r)
            Kall[(size_t)(mbase + r) * HDIM + n] = acc[r] + bias;
    }
}

// nrm[i] = sum_j Kall[i,j]^2;  grid=NR, block=256
__global__ void k_rownorm(const float* __restrict__ Kall, float* __restrict__ nrm) {
    __shared__ float sm[256];
    float v = Kall[(size_t)blockIdx.x * HDIM + threadIdx.x];
    float s = block_reduce_256(v * v, sm);
    if (threadIdx.x == 0) nrm[blockIdx.x] = s;
}

// Out[m,n] = exp(-sqrt(max(nrm_a + nrm_b - 2*dot, 0))) + B * (coords equal)
// One 16x64 strip per wave. grid = (Mrows/16)*(NSUP/64)/8, block=256
__global__ void k_gram_epi(const float* __restrict__ Kall, const float* __restrict__ nrm,
                           const float* __restrict__ S, const float* __restrict__ Qs,
                           const float* __restrict__ Bval, float* __restrict__ Out,
                           int Mrows, int rowA0, int isMix) {
    int lane = threadIdx.x & 31;
    int wave = threadIdx.x >> 5;
    int strip = blockIdx.x * 8 + wave;
    const int NG = NSUP / 64;                 // 32 N-groups
    int tm = strip / NG, ng = strip % NG;

    int arow = rowA0 + tm * 16 + (lane & 15);
    int brow = ng * 64 + (lane & 15);
    int khalf = (lane >> 4) * 2;
    const float* ap = Kall + (size_t)arow * HDIM + khalf;
    const float* bp = Kall + (size_t)brow * HDIM + khalf;

    v8f acc0 = {}, acc1 = {}, acc2 = {}, acc3 = {};
    for (int k0 = 0; k0 < HDIM; k0 += 4) {
        v2f a = *(const v2f*)(ap + k0);
        v2f b0 = *(const v2f*)(bp + k0);
        v2f b1 = *(const v2f*)(bp + k0 + 16 * HDIM);
        v2f b2 = *(const v2f*)(bp + k0 + 32 * HDIM);
        v2f b3 = *(const v2f*)(bp + k0 + 48 * HDIM);
        acc0 = wmma_f32_16x16x4(a, b0, acc0);
        acc1 = wmma_f32_16x16x4(a, b1, acc1);
        acc2 = wmma_f32_16x16x4(a, b2, acc2);
        acc3 = wmma_f32_16x16x4(a, b3, acc3);
    }
    float Bv = Bval[0];
    int mbase = tm * 16 + ((lane >> 4) << 3);
    float nrm_a[8], ax0[8], ax1[8];
#pragma unroll
    for (int r = 0; r < 8; ++r) {
        int m = mbase + r;
        nrm_a[r] = nrm[rowA0 + m];
        if (isMix) { ax0[r] = Qs[m * 2 + 0]; ax1[r] = Qs[m * 2 + 1]; }
        else       { ax0[r] = S[m * 3 + 0];  ax1[r] = S[m * 3 + 1];  }
    }
#pragma unroll
    for (int t = 0; t < 4; ++t) {
        v8f acc = (t == 0) ? acc0 : (t == 1) ? acc1 : (t == 2) ? acc2 : acc3;
        int n = ng * 64 + t * 16 + (lane & 15);
        float nrm_n = nrm[n];
        float bx0 = S[n * 3 + 0], bx1 = S[n * 3 + 1];
#pragma unroll
        for (int r = 0; r < 8; ++r) {
            int m = mbase + r;
            float g = nrm_a[r] + nrm_n - 2.f * acc[r];
            float d = sqrtf(fmaxf(g, 0.f));
            float eq = (ax0[r] == bx0 && ax1[r] == bx1) ? 1.f : 0.f;
            Out[(size_t)m * NSUP + n] = expf(-d) + Bv * eq;
        }
    }
}

// CG init: x=0, r=p=support_y - meanv;  grid=NSUP/256, block=256
__global__ void k_cg_init(const float* __restrict__ S, const float* __restrict__ meanv,
                          float* __restrict__ rv, float* __restrict__ pv,
                          float* __restrict__ xv) {
    int i = blockIdx.x * 256 + threadIdx.x;
    float r = S[i * 3 + 2] - meanv[i];
    rv[i] = r; pv[i] = r; xv[i] = 0.f;
}

// q = selfk @ p, with p staged to LDS via async copy;  grid=NSUP, block=256
__global__ void k_matvec_selfk(const float* __restrict__ A, const float* __restrict__ p,
                               float* __restrict__ q) {
    __shared__ float pl[NSUP];
    __shared__ float sm[256];
    int base = threadIdx.x * 8;
    async_copy_b128(&pl[base], p + base);
    async_copy_b128(&pl[base + 4], p + base + 4);
    async_wait_all();
    __syncthreads();
    int i = blockIdx.x;
    const float* row = A + (size_t)i * NSUP;
    float acc = 0.f;
    for (int j = threadIdx.x; j < NSUP; j += 256) acc = fmaf(row[j], pl[j], acc);
    float s = block_reduce_256(acc, sm);
    if (threadIdx.x == 0) q[i] = s;
}

// Full CG scalar step fused into one block:
// pq=p.q; rr=r.r; a=rr/pq; x+=a p; r-=a q; rrn=r.r; b=rrn/rr; p=r+b p
// grid=1, block=256 (each thread owns 8 elements)
__global__ void k_cg_step(const float* __restrict__ qv, float* __restrict__ pv,
                          float* __restrict__ rv, float* __restrict__ xv) {
    __shared__ float sm[256];
    float pq = 0.f, rr = 0.f;
    float pl[8], ql[8], rl[8];
#pragma unroll
    for (int u = 0; u < 8; ++u) {
        int i = threadIdx.x + 256 * u;
        pl[u] = pv[i]; ql[u] = qv[i]; rl[u] = rv[i];
        pq = fmaf(pl[u], ql[u], pq);
        rr = fmaf(rl[u], rl[u], rr);
    }
    pq = block_reduce_256(pq, sm);
    rr = block_reduce_256(rr, sm);
    float a = (fabsf(pq) > 1e-30f) ? rr / pq : 0.f;
    float rrn = 0.f;
#pragma unroll
    for (int u = 0; u < 8; ++u) {
        int i = threadIdx.x + 256 * u;
        xv[i] = fmaf(a, pl[u], xv[i]);
        rl[u] = fmaf(-a, ql[u], rl[u]);
        rv[i] = rl[u];
        rrn = fmaf(rl[u], rl[u], rrn);
    }
    rrn = block_reduce_256(rrn, sm);
    float b = (rr > 1e-30f) ? rrn / rr : 0.f;
#pragma unroll
    for (int u = 0; u < 8; ++u) {
        int i = threadIdx.x + 256 * u;
        pv[i] = fmaf(b, pl[u], rl[u]);
    }
}

// y[q] = meanv[NSUP+q] + mixk[q,:] . alpha (alpha staged to LDS async);  grid=NQRY
__global__ void k_final(const float* __restrict__ mixk, const float* __restrict__ xv,
                        const float* __restrict__ meanv, float* __restrict__ y) {
    __shared__ float xl[NSUP];
    __shared__ float sm[256];
    int base = threadIdx.x * 8;
    async_copy_b128(&xl[base], xv + base);
    async_copy_b128(&xl[base + 4], xv + base + 4);
    async_wait_all();
    __syncthreads();
    int q = blockIdx.x;
    const float* row = mixk + (size_t)q * NSUP;
    float acc = 0.f;
    for (int j = threadIdx.x; j < NSUP; j += 256) acc = fmaf(row[j], xl[j], acc);
    float s = block_reduce_256(acc, sm);
    if (threadIdx.x == 0) y[q] = meanv[NSUP + q] + s;
}

extern "C" void kernel_launch(void* const* d_in, const int* in_sizes, int n_in,
                              void* d_out, int out_size, void* d_ws, size_t ws_size,
                              hipStream_t stream) {
    (void)in_sizes; (void)n_in; (void)out_size; (void)ws_size;
    const float* S   = (const float*)d_in[0];
    const float* Qs  = (const float*)d_in[1];
    const float* Wz1 = (const float*)d_in[2];
    const float* bz1 = (const float*)d_in[3];
    const float* Wz2 = (const float*)d_in[4];
    const float* bz2 = (const float*)d_in[5];
    const float* Wb1 = (const float*)d_in[6];
    const float* bb1 = (const float*)d_in[7];
    const float* Wb2 = (const float*)d_in[8];
    const float* bb2 = (const float*)d_in[9];
    const float* Wm1 = (const float*)d_in[10];
    const float* bm1 = (const float*)d_in[11];
    const float* Wm2 = (const float*)d_in[12];
    const float* bm2 = (const float*)d_in[13];
    const float* Wk1 = (const float*)d_in[14];
    const float* bk1 = (const float*)d_in[15];
    const float* Wk2 = (const float*)d_in[16];
    const float* bk2 = (const float*)d_in[17];

    float* ws = (float*)d_ws;
    size_t o = 0;
    float* Hk    = ws + o; o += (size_t)NR * HDIM;
    float* Kall  = ws + o; o += (size_t)NR * HDIM;
    float* selfk = ws + o; o += (size_t)NSUP * NSUP;
    float* mixk  = ws + o; o += (size_t)NQRY * NSUP;
    float* hbar  = ws + o; o += HDIM;
    float* Zv    = ws + o; o += HDIM;
    float* cm    = ws + o; o += HDIM;
    float* ck    = ws + o; o += HDIM;
    float* hb    = ws + o; o += HDIM;
    float* Bv    = ws + o; o += 8;
    float* meanv = ws + o; o += NR;
    float* nrm   = ws + o; o += NR;
    float* rv    = ws + o; o += NSUP;
    float* pv    = ws + o; o += NSUP;
    float* qv    = ws + o; o += NSUP;
    float* xv    = ws + o; o += NSUP;

    // Z-network (mean commutes with linear layer 2)
    k_colmean_h1<<<HDIM, 256, 0, stream>>>(S, Wz1, bz1, hbar);
    k_matvec<<<HDIM, 256, 0, stream>>>(Wz2, HDIM, 0, hbar, bz2, Zv, HDIM, 0);
    // constant parts of mean/K first layers; B-net
    k_matvec<<<HDIM, 256, 0, stream>>>(Wm1, HDIM + 2, 2, Zv, bm1, cm, HDIM, 0);
    k_matvec<<<HDIM, 256, 0, stream>>>(Wk1, HDIM + 2, 2, Zv, bk1, ck, HDIM, 0);
    k_matvec<<<HDIM, 256, 0, stream>>>(Wb1, HDIM, 0, Zv, bb1, hb, HDIM, 1);
    k_dot_bias<<<1, 256, 0, stream>>>(Wb2, hb, bb2, Bv);
    // first hidden layers + mean heads
    k_layer1<<<NR, 256, 0, stream>>>(S, Qs, Wm1, cm, Wk1, ck, Wm2, bm2, Hk, meanv);
    // K second layer (WMMA fp32, 16x64 strips)
    k_gemm_kall<<<(NR / 16) * (HDIM / 64) / 8, 256, 0, stream>>>(Hk, Wk2, bk2, Kall);
    // kernel matrices (WMMA fp32 Gram + fused epilogue)
    k_rownorm<<<NR, 256, 0, stream>>>(Kall, nrm);
    k_gram_epi<<<(NSUP / 16) * (NSUP / 64) / 8, 256, 0, stream>>>(
        Kall, nrm, S, Qs, Bv, selfk, NSUP, 0, 0);
    k_gram_epi<<<(NQRY / 16) * (NSUP / 64) / 8, 256, 0, stream>>>(
        Kall, nrm, S, Qs, Bv, mixk, NQRY, NSUP, 1);
    // CG solve: selfk @ alpha = support_y - support_mean (2 launches / iter)
    k_cg_init<<<NSUP / 256, 256, 0, stream>>>(S, meanv, rv, pv, xv);
    for (int it = 0; it < CG_ITERS; ++it) {
        k_matvec_selfk<<<NSUP, 256, 0, stream>>>(selfk, pv, qv);
        k_cg_step<<<1, 256, 0, stream>>>(qv, pv, rv, xv);
    }
    // y = query_mean + mix_k @ alpha
    k_final<<<NQRY, 256, 0, stream>>>(mixk, xv, meanv, (float*)d_out);
}